// Encoder_67808943669372
// MI455X (gfx1250) — compile-verified
//
#include <hip/hip_runtime.h>
#include <hip/hip_bf16.h>
#include <math.h>

typedef __attribute__((ext_vector_type(2))) float v2f;
typedef __attribute__((ext_vector_type(8))) float v8f;

#define D_FEAT 128
#define MT_PER_BLK 2           // 16-row M-tiles per block -> 32 rows/block
#define ROWS_PER_BLK (MT_PER_BLK * 16)

// ---------------------------------------------------------------------------
// 1) degree accumulation: deg[dst] += 1 per edge (self-loop added in dinv pass)
// ---------------------------------------------------------------------------
__global__ __launch_bounds__(256) void deg_kernel(const int* __restrict__ dst,
                                                  float* __restrict__ deg,
                                                  int n_edges) {
    int e = blockIdx.x * blockDim.x + threadIdx.x;
    if (e < n_edges) atomicAdd(&deg[dst[e]], 1.0f);
}

__global__ __launch_bounds__(256) void dinv_kernel(float* __restrict__ deg, int n) {
    int i = blockIdx.x * blockDim.x + threadIdx.x;
    if (i < n) deg[i] = rsqrtf(deg[i] + 1.0f);   // deg includes self-loop -> >= 1
}

// ---------------------------------------------------------------------------
// 2) h' = (x @ W) * dinv[row], written to hprime AND acc (acc starts with the
//    self-loop contribution).  WMMA f32 16x16x4, wave32.
//    A 16x4 layout: lanes 0-15 row M=lane, K = k0+{0,1}; lanes 16-31 K = k0+{2,3}
//    B 4x16 layout: VGPR v -> row K = k0 + v + (lane>=16 ? 2 : 0), col N = lane&15
//    C/D: VGPR r -> row M = r + (lane>=16 ? 8 : 0), col N = lane&15
// ---------------------------------------------------------------------------
__global__ __launch_bounds__(256) void gemm_scale_kernel(
    const float* __restrict__ x, const float* __restrict__ W,
    const float* __restrict__ dinv, float* __restrict__ hprime,
    float* __restrict__ acc, int n_nodes) {
    __shared__ float sX[ROWS_PER_BLK][D_FEAT + 4];   // +4 pad: conflict-free, 16B rows

    const int tid   = threadIdx.x;
    const int lane  = tid & 31;
    const int wave  = tid >> 5;            // 0..7 -> N-tile
    const int mbase = blockIdx.x * ROWS_PER_BLK;
    const int half  = lane >> 4;           // 0 | 1
    const int l15   = lane & 15;
    const int nb    = wave * 16;

    // block-uniform full-tile predicate: single scalar branch, no per-elem guards
    const bool full = (mbase + ROWS_PER_BLK) <= n_nodes;

    // cooperative stage of 32x128 x-tile: 1024 float4 slots, 4 per thread
    const float4* xv = reinterpret_cast<const float4*>(x);
    if (full) {
#pragma unroll
        for (int i = 0; i < 4; ++i) {
            int slot = tid + i * 256;          // 0..1023
            int r    = slot >> 5;              // row in tile
            int cb   = slot & 31;              // float4 column block
            *reinterpret_cast<float4*>(&sX[r][cb * 4]) =
                xv[(size_t)(mbase + r) * (D_FEAT / 4) + cb];
        }
    } else {
        const float4 zero4 = make_float4(0.f, 0.f, 0.f, 0.f);
#pragma unroll
        for (int i = 0; i < 4; ++i) {
            int slot = tid + i * 256;
            int r    = slot >> 5;
            int cb   = slot & 31;
            int row  = mbase + r;
            float4 v = (row < n_nodes) ? xv[(size_t)row * (D_FEAT / 4) + cb] : zero4;
            *reinterpret_cast<float4*>(&sX[r][cb * 4]) = v;
        }
    }
    __syncthreads();

    v8f c[MT_PER_BLK];
#pragma unroll
    for (int t = 0; t < MT_PER_BLK; ++t)
#pragma unroll
        for (int r = 0; r < 8; ++r) c[t][r] = 0.0f;

    for (int k0 = 0; k0 < D_FEAT; k0 += 4) {
        const int ka = k0 + half * 2;
        v2f bf;
        bf.x = W[(size_t)ka * D_FEAT + nb + l15];
        bf.y = W[(size_t)(ka + 1) * D_FEAT + nb + l15];
#pragma unroll
        for (int t = 0; t < MT_PER_BLK; ++t) {
            v2f af;
            af.x = sX[t * 16 + l15][ka];
            af.y = sX[t * 16 + l15][ka + 1];
            c[t] = __builtin_amdgcn_wmma_f32_16x16x4_f32(
                /*neg_a=*/false, af, /*neg_b=*/false, bf,
                /*c_mod=*/(short)0, c[t], /*reuse_a=*/false, /*reuse_b=*/false);
        }
    }

    // ---- epilogue: scale by dinv[row], dual store (hprime + acc seed) ----
    if (full) {
#pragma unroll
        for (int t = 0; t < MT_PER_BLK; ++t) {
            const int rbase = mbase + t * 16 + half * 8;     // row of c[t][0]
            float* hp = hprime + (size_t)rbase * D_FEAT + nb + l15;
            float* ap = acc    + (size_t)rbase * D_FEAT + nb + l15;
            float dv[8];
#pragma unroll
            for (int r = 0; r < 8; ++r) dv[r] = dinv[rbase + r];
#pragma unroll
            for (int r = 0; r < 8; ++r) {
                float v = c[t][r] * dv[r];
                hp[(size_t)r * D_FEAT] = v;
                ap[(size_t)r * D_FEAT] = v;
            }
        }
    } else {
#pragma unroll
        for (int t = 0; t < MT_PER_BLK; ++t) {
#pragma unroll
            for (int r = 0; r < 8; ++r) {
                int row = mbase + t * 16 + r + half * 8;
                if (row < n_nodes) {
                    float v = c[t][r] * dinv[row];
                    size_t off = (size_t)row * D_FEAT + nb + l15;
                    hprime[off] = v;
                    acc[off]    = v;
                }
            }
        }
    }
}

// ---------------------------------------------------------------------------
// 3) edge scatter: acc[dst] += h'[src]  (one wave per edge, float4 per lane)
//    h' (51 MB) is L2-resident on MI455X (192 MB L2) -> random gather stays on-die
// ---------------------------------------------------------------------------
__global__ __launch_bounds__(256) void scatter_kernel(
    const int* __restrict__ src, const int* __restrict__ dst,
    const float* __restrict__ hprime, float* __restrict__ acc, int n_edges) {
    long long gid = (long long)blockIdx.x * blockDim.x + threadIdx.x;
    int e    = (int)(gid >> 5);
    int lane = (int)(gid & 31);
    if (e >= n_edges) return;
    int s = src[e], d = dst[e];
    float4 hv = *reinterpret_cast<const float4*>(hprime + (size_t)s * D_FEAT + lane * 4);
    float* ap = acc + (size_t)d * D_FEAT + lane * 4;
    atomicAdd(ap + 0, hv.x);
    atomicAdd(ap + 1, hv.y);
    atomicAdd(ap + 2, hv.z);
    atomicAdd(ap + 3, hv.w);
}

// ---------------------------------------------------------------------------
// 4) finalize: out = dinv*acc + b ; PReLU ; L2-normalize (wave32 butterfly) ;
//    atomic accumulate into pooled[batch[i]] and per-graph count
// ---------------------------------------------------------------------------
__global__ __launch_bounds__(256) void finalize_kernel(
    const float* __restrict__ acc, const float* __restrict__ dinv,
    const float* __restrict__ bias, const float* __restrict__ pa,
    const int* __restrict__ batch, float* __restrict__ pooled,
    float* __restrict__ cnt, int n_nodes) {
    long long gid = (long long)blockIdx.x * blockDim.x + threadIdx.x;
    int i    = (int)(gid >> 5);
    int lane = (int)(gid & 31);
    if (i >= n_nodes) return;

    float di = dinv[i];
    float v[4];
    float ss = 0.0f;
#pragma unroll
    for (int j = 0; j < 4; ++j) {
        int c   = lane + 32 * j;
        float t = di * acc[(size_t)i * D_FEAT + c] + bias[c];
        t = (t > 0.0f) ? t : pa[c] * t;
        v[j] = t;
        ss += t * t;
    }
#pragma unroll
    for (int off = 16; off > 0; off >>= 1) ss += __shfl_xor(ss, off, 32);
    float scale = 1.0f / fmaxf(sqrtf(ss), 1e-12f);

    int g = batch[i];
#pragma unroll
    for (int j = 0; j < 4; ++j)
        atomicAdd(&pooled[(size_t)g * D_FEAT + lane + 32 * j], v[j] * scale);
    if (lane == 0) atomicAdd(&cnt[g], 1.0f);
}

// ---------------------------------------------------------------------------
// 5) pooled /= max(cnt, 1)
// ---------------------------------------------------------------------------
__global__ __launch_bounds__(256) void div_kernel(float* __restrict__ pooled,
                                                  const float* __restrict__ cnt,
                                                  int total) {
    int i = blockIdx.x * blockDim.x + threadIdx.x;
    if (i < total) pooled[i] /= fmaxf(cnt[i >> 7], 1.0f);
}

// ---------------------------------------------------------------------------
extern "C" void kernel_launch(void* const* d_in, const int* in_sizes, int n_in,
                              void* d_out, int out_size, void* d_ws, size_t ws_size,
                              hipStream_t stream) {
    const float* x     = (const float*)d_in[0];
    const int*   ei    = (const int*)d_in[1];
    const int*   batch = (const int*)d_in[2];
    const float* W     = (const float*)d_in[3];
    const float* bias  = (const float*)d_in[4];
    const float* pa    = (const float*)d_in[5];

    const int n_nodes = in_sizes[0] / D_FEAT;
    const int n_edges = in_sizes[1] / 2;
    const int* src = ei;
    const int* dst = ei + n_edges;
    float* pooled = (float*)d_out;

    // workspace carve-out: hprime[N*128] | acc[N*128] | dinv[N] | cnt[128]
    float* hprime = (float*)d_ws;
    float* acc    = hprime + (size_t)n_nodes * D_FEAT;
    float* dinv   = acc + (size_t)n_nodes * D_FEAT;
    float* cnt    = dinv + n_nodes;

    hipMemsetAsync(dinv, 0, (size_t)n_nodes * sizeof(float), stream);
    hipMemsetAsync(cnt, 0, 128 * sizeof(float), stream);
    hipMemsetAsync(d_out, 0, (size_t)out_size * sizeof(float), stream);

    deg_kernel<<<(n_edges + 255) / 256, 256, 0, stream>>>(dst, dinv, n_edges);
    dinv_kernel<<<(n_nodes + 255) / 256, 256, 0, stream>>>(dinv, n_nodes);

    int gemm_blocks = (n_nodes + ROWS_PER_BLK - 1) / ROWS_PER_BLK;
    gemm_scale_kernel<<<gemm_blocks, 256, 0, stream>>>(x, W, dinv, hprime, acc, n_nodes);

    long long sc_threads = (long long)n_edges * 32;
    scatter_kernel<<<(int)((sc_threads + 255) / 256), 256, 0, stream>>>(src, dst, hprime, acc, n_edges);

    long long fin_threads = (long long)n_nodes * 32;
    finalize_kernel<<<(int)((fin_threads + 255) / 256), 256, 0, stream>>>(
        acc, dinv, bias, pa, batch, pooled, cnt, n_nodes);

    div_kernel<<<(out_size + 255) / 256, 256, 0, stream>>>(pooled, cnt, out_size);
}